// ResGATv2Block_6571299963426
// MI455X (gfx1250) — compile-verified
//
#include <hip/hip_runtime.h>
#include <hip/hip_bf16.h>

#define N_    50000
#define E_    1600000
#define ETOT_ (E_ + N_)
#define CH_   128
#define H_    8
#define EPS_  1e-5f
#define SLOPE_ 0.2f

typedef __attribute__((ext_vector_type(2))) float v2f;
typedef __attribute__((ext_vector_type(8))) float v8f;

// order-preserving int key for float atomicMax
__device__ __forceinline__ int fkey(float f) {
  int b = __float_as_int(f);
  return b >= 0 ? b : (b ^ 0x7fffffff);
}
__device__ __forceinline__ float funkey(int k) {
  return __int_as_float(k >= 0 ? k : (k ^ 0x7fffffff));
}
__device__ __forceinline__ float lrelu(float v) { return v > 0.f ? v : SLOPE_ * v; }

// ---------------- fills ----------------
__global__ __launch_bounds__(256) void fill_f32(float* p, float v, long long n) {
  long long i = (long long)blockIdx.x * 256 + threadIdx.x;
  if (i < n) p[i] = v;
}
__global__ __launch_bounds__(256) void fill_i32(int* p, int v, long long n) {
  long long i = (long long)blockIdx.x * 256 + threadIdx.x;
  if (i < n) p[i] = v;
}

// ---------------- self-loop attr (mean of edge_attr over dst) ----------------
__global__ __launch_bounds__(256) void loop_sum_kernel(const int* __restrict__ dstA,
                                                       const float* __restrict__ eattr,
                                                       float* __restrict__ sums,
                                                       int* __restrict__ cnt) {
  long long t = (long long)blockIdx.x * 256 + threadIdx.x;   // e*16 + k
  if (t >= (long long)E_ * 16) return;
  int e = (int)(t >> 4), k = (int)(t & 15);
  int d = dstA[e];
  atomicAdd(&sums[(long long)d * 16 + k], eattr[t]);
  if (k == 0) atomicAdd(&cnt[d], 1);
}
__global__ __launch_bounds__(256) void loop_mean_kernel(const int* __restrict__ cnt,
                                                        float* __restrict__ sums) {
  long long i = (long long)blockIdx.x * 256 + threadIdx.x;   // n*16 + k
  if (i >= (long long)N_ * 16) return;
  int c = cnt[i >> 4];
  sums[i] = (c > 0) ? sums[i] / (float)c : 0.f;
}

// ---------------- fused dual projection with f32 WMMA ----------------
// xl = xin @ wl + bl ; xr = xin @ wr + br   (one 16x16 tile per wave, K=128)
__global__ __launch_bounds__(256) void proj_wmma(const float* __restrict__ xin,
                                                 const float* __restrict__ wl,
                                                 const float* __restrict__ bl,
                                                 const float* __restrict__ wr,
                                                 const float* __restrict__ br,
                                                 float* __restrict__ xl,
                                                 float* __restrict__ xr) {
  const int lane = threadIdx.x & 31;
  const int wave = threadIdx.x >> 5;     // 8 waves -> 8 N-tiles (all 128 cols)
  const int m0 = blockIdx.x * 16;
  const int n0 = wave * 16;
  const int half = lane >> 4;
  const int r = lane & 15;

  v8f accL = {};
  v8f accR = {};
  const long long rowA = (long long)(m0 + r) * CH_;
  for (int kb = 0; kb < 32; ++kb) {
    const int k0 = kb * 4;
    const int ka0 = k0 + (half ? 2 : 0);
    const int ka1 = k0 + (half ? 3 : 1);
    v2f a;  a.x  = xin[rowA + ka0];          a.y  = xin[rowA + ka1];
    v2f b0; b0.x = wl[ka0 * CH_ + n0 + r];   b0.y = wl[ka1 * CH_ + n0 + r];
    v2f b1; b1.x = wr[ka0 * CH_ + n0 + r];   b1.y = wr[ka1 * CH_ + n0 + r];
    accL = __builtin_amdgcn_wmma_f32_16x16x4_f32(false, a, false, b0, (short)0, accL, false, false);
    accR = __builtin_amdgcn_wmma_f32_16x16x4_f32(false, a, false, b1, (short)0, accR, false, false);
  }
  const int col = n0 + r;
  const float bL = bl[col], bR = br[col];
#pragma unroll
  for (int v = 0; v < 8; ++v) {
    long long row = m0 + v + half * 8;
    xl[row * CH_ + col] = accL[v] + bL;
    xr[row * CH_ + col] = accR[v] + bR;
  }
}

// ---------------- edge pass A: logits + segment max ----------------
// one wave per edge; lane owns 4 consecutive channels; head = lane>>2
__global__ __launch_bounds__(256) void edge_alpha(const int* __restrict__ srcA,
                                                  const int* __restrict__ dstA,
                                                  const float* __restrict__ eattr,
                                                  const float* __restrict__ loopattr,
                                                  const float* __restrict__ xl,
                                                  const float* __restrict__ xr,
                                                  const float* __restrict__ we,
                                                  const float* __restrict__ att,
                                                  float* __restrict__ alphaBuf,
                                                  int* __restrict__ amaxKey) {
  const int lane = threadIdx.x & 31;
  const long long e = (long long)blockIdx.x * 8 + (threadIdx.x >> 5);
  if (e >= ETOT_) return;
  int s, d;
  const float* ep;
  if (e < E_) { s = srcA[e]; d = dstA[e]; ep = eattr + e * 16; }
  else        { s = d = (int)(e - E_);    ep = loopattr + (long long)s * 16; }

  float eav = (lane < 16) ? ep[lane] : 0.f;
  const int ch = lane * 4;
  float4 ee = make_float4(0.f, 0.f, 0.f, 0.f);
#pragma unroll
  for (int k = 0; k < 16; ++k) {
    float ak = __shfl(eav, k, 32);
    const float4 w4 = *(const float4*)(we + k * CH_ + ch);
    ee.x += ak * w4.x; ee.y += ak * w4.y; ee.z += ak * w4.z; ee.w += ak * w4.w;
  }
  const float4 xlv = *(const float4*)(xl + (long long)s * CH_ + ch);
  const float4 xrv = *(const float4*)(xr + (long long)d * CH_ + ch);
  float4 m;
  m.x = lrelu(xlv.x + xrv.x + ee.x);
  m.y = lrelu(xlv.y + xrv.y + ee.y);
  m.z = lrelu(xlv.z + xrv.z + ee.z);
  m.w = lrelu(xlv.w + xrv.w + ee.w);
  const float4 a4 = *(const float4*)(att + (lane >> 2) * 16 + (lane & 3) * 4);
  float p = m.x * a4.x + m.y * a4.y + m.z * a4.z + m.w * a4.w;
  p += __shfl_xor(p, 1, 32);
  p += __shfl_xor(p, 2, 32);
  if ((lane & 3) == 0) {
    int h = lane >> 2;
    alphaBuf[e * 8 + h] = p;
    atomicMax(&amaxKey[(long long)d * 8 + h], fkey(p));
  }
}

// ---------------- edge pass B: exp + segment sum ----------------
__global__ __launch_bounds__(256) void edge_exp(const int* __restrict__ dstA,
                                                const int* __restrict__ amaxKey,
                                                float* __restrict__ alphaBuf,
                                                float* __restrict__ denom) {
  long long t = (long long)blockIdx.x * 256 + threadIdx.x;   // e*8 + h
  if (t >= (long long)ETOT_ * 8) return;
  long long e = t >> 3;
  int h = (int)(t & 7);
  int d = (e < E_) ? dstA[e] : (int)(e - E_);
  float am = funkey(amaxKey[(long long)d * 8 + h]);
  float ex = expf(alphaBuf[t] - am);
  alphaBuf[t] = ex;
  atomicAdd(&denom[(long long)d * 8 + h], ex);
}

// ---------------- edge pass C: weighted scatter aggregation ----------------
__global__ __launch_bounds__(256) void edge_aggr(const int* __restrict__ srcA,
                                                 const int* __restrict__ dstA,
                                                 const float* __restrict__ xl,
                                                 const float* __restrict__ exBuf,
                                                 const float* __restrict__ denom,
                                                 float* __restrict__ outAcc) {
  const int lane = threadIdx.x & 31;
  const long long e = (long long)blockIdx.x * 8 + (threadIdx.x >> 5);
  if (e >= ETOT_) return;
  int s, d;
  if (e < E_) { s = srcA[e]; d = dstA[e]; }
  else        { s = d = (int)(e - E_); }
  const int h = lane >> 2;
  const float dn = denom[(long long)d * 8 + h];
  const float a = exBuf[e * 8 + h] / fmaxf(dn, 1e-16f);
  const int ch = lane * 4;
  const float4 xlv = *(const float4*)(xl + (long long)s * CH_ + ch);
  float* op = outAcc + (long long)d * CH_ + ch;
  atomicAdd(op + 0, xlv.x * a);
  atomicAdd(op + 1, xlv.y * a);
  atomicAdd(op + 2, xlv.z * a);
  atomicAdd(op + 3, xlv.w * a);
}

// ---------------- BN stats: per-channel sum / sumsq ----------------
__global__ __launch_bounds__(256) void bn_stats(const float* __restrict__ h,
                                                float* __restrict__ stats) {
  __shared__ float s1[CH_], s2[CH_];
  if (threadIdx.x < CH_) { s1[threadIdx.x] = 0.f; s2[threadIdx.x] = 0.f; }
  __syncthreads();
  const long long total = (long long)N_ * CH_;
  for (long long i = (long long)blockIdx.x * blockDim.x + threadIdx.x; i < total;
       i += (long long)gridDim.x * blockDim.x) {
    float v = h[i];
    int ch = (int)(i & (CH_ - 1));
    atomicAdd(&s1[ch], v);
    atomicAdd(&s2[ch], v * v);
  }
  __syncthreads();
  if (threadIdx.x < CH_) {
    atomicAdd(&stats[threadIdx.x], s1[threadIdx.x]);
    atomicAdd(&stats[CH_ + threadIdx.x], s2[threadIdx.x]);
  }
}

// ---------------- BN normalize (+optional residual) + ReLU ----------------
__global__ __launch_bounds__(256) void bn_finalize(const float* __restrict__ h,
                                                   const float* __restrict__ stats,
                                                   const float* __restrict__ gamma,
                                                   const float* __restrict__ beta,
                                                   const float* __restrict__ resid,
                                                   float* __restrict__ out) {
  long long i = (long long)blockIdx.x * 256 + threadIdx.x;
  if (i >= (long long)N_ * CH_) return;
  int ch = (int)(i & (CH_ - 1));
  const float invN = 1.f / (float)N_;
  float mean = stats[ch] * invN;
  float var = stats[CH_ + ch] * invN - mean * mean;
  float y = gamma[ch] * (h[i] - mean) * rsqrtf(var + EPS_) + beta[ch];
  if (resid) y += resid[i];
  out[i] = fmaxf(y, 0.f);
}

static inline unsigned cdiv(long long a, long long b) { return (unsigned)((a + b - 1) / b); }

extern "C" void kernel_launch(void* const* d_in, const int* in_sizes, int n_in,
                              void* d_out, int out_size, void* d_ws, size_t ws_size,
                              hipStream_t stream) {
  (void)in_sizes; (void)n_in; (void)out_size; (void)ws_size;
  const float* x     = (const float*)d_in[0];
  const int*   ei    = (const int*)d_in[1];
  const float* eattr = (const float*)d_in[2];
  const int* srcA = ei;
  const int* dstA = ei + E_;

  // per-layer params: wl, bl, wr, br, we, att, gamma, beta (c*_bias cancels in BN)
  const float* P[2][8];
  for (int l = 0; l < 2; ++l) {
    int b = 3 + l * 9;
    P[l][0] = (const float*)d_in[b + 0];  // wl
    P[l][1] = (const float*)d_in[b + 1];  // bl
    P[l][2] = (const float*)d_in[b + 2];  // wr
    P[l][3] = (const float*)d_in[b + 3];  // br
    P[l][4] = (const float*)d_in[b + 4];  // we
    P[l][5] = (const float*)d_in[b + 5];  // att
    P[l][6] = (const float*)d_in[b + 7];  // gamma
    P[l][7] = (const float*)d_in[b + 8];  // beta
  }

  // workspace layout
  char* ws = (char*)d_ws;
  float* xl       = (float*)ws; ws += (size_t)N_ * CH_ * 4;
  float* xr       = (float*)ws; ws += (size_t)N_ * CH_ * 4;
  float* hpre     = (float*)ws; ws += (size_t)N_ * CH_ * 4;
  float* exbuf    = (float*)ws; ws += (size_t)ETOT_ * H_ * 4;
  int*   amaxKey  = (int*)  ws; ws += (size_t)N_ * H_ * 4;
  float* denom    = (float*)ws; ws += (size_t)N_ * H_ * 4;
  float* loopattr = (float*)ws; ws += (size_t)N_ * 16 * 4;
  int*   cnt      = (int*)  ws; ws += (size_t)N_ * 4;
  float* stats    = (float*)ws; ws += 256 * 4;

  // ---- self-loop edge_attr (mean over incoming edges) ----
  fill_f32<<<cdiv((long long)N_ * 16, 256), 256, 0, stream>>>(loopattr, 0.f, (long long)N_ * 16);
  fill_i32<<<cdiv(N_, 256), 256, 0, stream>>>(cnt, 0, N_);
  loop_sum_kernel<<<cdiv((long long)E_ * 16, 256), 256, 0, stream>>>(dstA, eattr, loopattr, cnt);
  loop_mean_kernel<<<cdiv((long long)N_ * 16, 256), 256, 0, stream>>>(cnt, loopattr);

  const int NEGINF_KEY = (int)0x807FFFFFu;  // fkey(-inf)

  for (int l = 0; l < 2; ++l) {
    const float* xin = (l == 0) ? x : (const float*)d_out;
    fill_i32<<<cdiv((long long)N_ * H_, 256), 256, 0, stream>>>(amaxKey, NEGINF_KEY, (long long)N_ * H_);
    fill_f32<<<cdiv((long long)N_ * H_, 256), 256, 0, stream>>>(denom, 0.f, (long long)N_ * H_);
    fill_f32<<<cdiv((long long)N_ * CH_, 256), 256, 0, stream>>>(hpre, 0.f, (long long)N_ * CH_);
    fill_f32<<<1, 256, 0, stream>>>(stats, 0.f, 256);

    proj_wmma<<<N_ / 16, 256, 0, stream>>>(xin, P[l][0], P[l][1], P[l][2], P[l][3], xl, xr);
    edge_alpha<<<cdiv(ETOT_, 8), 256, 0, stream>>>(srcA, dstA, eattr, loopattr, xl, xr,
                                                   P[l][4], P[l][5], exbuf, amaxKey);
    edge_exp<<<cdiv((long long)ETOT_ * 8, 256), 256, 0, stream>>>(dstA, amaxKey, exbuf, denom);
    edge_aggr<<<cdiv(ETOT_, 8), 256, 0, stream>>>(srcA, dstA, xl, exbuf, denom, hpre);

    bn_stats<<<512, 256, 0, stream>>>(hpre, stats);
    bn_finalize<<<cdiv((long long)N_ * CH_, 256), 256, 0, stream>>>(
        hpre, stats, P[l][6], P[l][7], (l == 1) ? x : nullptr, (float*)d_out);
  }
}